// ProposalLayer_33749853012640
// MI455X (gfx1250) — compile-verified
//
#include <hip/hip_runtime.h>
#include <hip/hip_bf16.h>

#define B_       4
#define PRE_NMS  2048
#define NPROP    1000
#define NMS_THR  0.7f
#define NBINS    4096
#define CAND_CAP 4096

typedef float v2f __attribute__((ext_vector_type(2)));
typedef float v8f __attribute__((ext_vector_type(8)));

// ---------------------------------------------------------------- zeroing
__global__ void zero_ws(unsigned int* p, int n) {
    int i = blockIdx.x * blockDim.x + threadIdx.x;
    int stride = gridDim.x * blockDim.x;
    for (; i < n; i += stride) p[i] = 0u;
}

// ---------------------------------------------------------------- 1) histogram of scores (LDS-privatized)
__global__ void hist_kernel(const float* __restrict__ probs,
                            unsigned int* __restrict__ hist, int N) {
    __shared__ unsigned int lh[NBINS];
    const int b = blockIdx.y;
    for (int i = threadIdx.x; i < NBINS; i += blockDim.x) lh[i] = 0u;
    __syncthreads();
    const float2* sp = (const float2*)(probs + (size_t)b * N * 2);
    const int stride = gridDim.x * blockDim.x;
    for (int i = blockIdx.x * blockDim.x + threadIdx.x; i < N; i += stride) {
        __builtin_prefetch(&sp[i + stride], 0, 1);  // global_prefetch_b8
        float s = sp[i].y;
        int bin = (int)(s * (float)NBINS);
        bin = bin < 0 ? 0 : (bin > NBINS - 1 ? NBINS - 1 : bin);
        atomicAdd(&lh[bin], 1u);
    }
    __syncthreads();
    for (int i = threadIdx.x; i < NBINS; i += blockDim.x)
        if (lh[i]) atomicAdd(&hist[b * NBINS + i], lh[i]);
}

// ---------------------------------------------------------------- 2) pick threshold bin per batch
__global__ void thresh_kernel(const unsigned int* __restrict__ hist,
                              int* __restrict__ tbin) {
    int b = threadIdx.x;
    if (b < B_) {
        unsigned int cum = 0; int t = 0;
        for (int i = NBINS - 1; i >= 0; --i) {
            cum += hist[b * NBINS + i];
            if (cum >= PRE_NMS) { t = i; break; }
        }
        tbin[b] = t;
    }
}

// ---------------------------------------------------------------- 3) compact candidates above threshold
__global__ void compact_kernel(const float* __restrict__ probs,
                               const int* __restrict__ tbin,
                               float* __restrict__ cs, int* __restrict__ ci,
                               unsigned int* __restrict__ cnt, int N) {
    const int b = blockIdx.y;
    const int t = tbin[b];
    const float2* sp = (const float2*)(probs + (size_t)b * N * 2);
    const int stride = gridDim.x * blockDim.x;
    for (int i = blockIdx.x * blockDim.x + threadIdx.x; i < N; i += stride) {
        float s = sp[i].y;
        int bin = (int)(s * (float)NBINS);
        bin = bin < 0 ? 0 : (bin > NBINS - 1 ? NBINS - 1 : bin);
        if (bin >= t) {
            unsigned int pos = atomicAdd(&cnt[b], 1u);
            if (pos < CAND_CAP) {
                cs[b * CAND_CAP + pos] = s;
                ci[b * CAND_CAP + pos] = i;
            }
        }
    }
}

// ---------------------------------------------------------------- 4) bitonic sort (desc) + build clipped boxes
__global__ void __launch_bounds__(1024)
sort_build_kernel(const float* __restrict__ cs, const int* __restrict__ ci,
                  const unsigned int* __restrict__ cnt,
                  const float* __restrict__ bbox, const float* __restrict__ anchors,
                  float* __restrict__ boxes, int N) {
    __shared__ float ss[CAND_CAP];
    __shared__ int   si[CAND_CAP];
    const int b = blockIdx.x;
    unsigned int n = cnt[b]; if (n > CAND_CAP) n = CAND_CAP;
    for (int k = threadIdx.x; k < CAND_CAP; k += blockDim.x) {
        if (k < (int)n) { ss[k] = cs[b * CAND_CAP + k]; si[k] = ci[b * CAND_CAP + k]; }
        else            { ss[k] = -1.0e30f;             si[k] = 0x7fffffff; }
    }
    __syncthreads();
    for (unsigned int k = 2; k <= CAND_CAP; k <<= 1) {
        for (unsigned int j = k >> 1; j > 0; j >>= 1) {
            for (unsigned int t = threadIdx.x; t < CAND_CAP / 2; t += blockDim.x) {
                unsigned int i = ((t & ~(j - 1)) << 1) | (t & (j - 1));
                unsigned int p = i | j;
                float s0 = ss[i], s1 = ss[p];
                int   i0 = si[i], i1 = si[p];
                // "p before i" in final (desc score, asc index) order:
                bool beforePI = (s1 > s0) || (s1 == s0 && i1 < i0);
                bool desc = ((i & k) == 0);
                if (beforePI == desc) { ss[i] = s1; ss[p] = s0; si[i] = i1; si[p] = i0; }
            }
            __syncthreads();
        }
    }
    for (int kq = threadIdx.x; kq < PRE_NMS; kq += blockDim.x) {
        int idx = si[kq];
        float4 box = make_float4(0.f, 0.f, 0.f, 0.f);
        if (idx >= 0 && idx < N) {
            const float4 a = ((const float4*)anchors)[(size_t)b * N + idx];
            const float4 d = ((const float4*)bbox)[(size_t)b * N + idx];
            float h = a.z - a.x, w = a.w - a.y;
            float dy = d.x * 0.1f, dx = d.y * 0.1f;
            float dh = d.z * 0.2f, dw = d.w * 0.2f;
            float cy = a.x + 0.5f * h + dy * h;
            float cx = a.y + 0.5f * w + dx * w;
            h *= __expf(dh); w *= __expf(dw);
            box.x = fminf(fmaxf(cy - 0.5f * h, 0.f), 1.f);
            box.y = fminf(fmaxf(cx - 0.5f * w, 0.f), 1.f);
            box.z = fminf(fmaxf(cy + 0.5f * h, 0.f), 1.f);
            box.w = fminf(fmaxf(cx + 0.5f * w, 0.f), 1.f);
        }
        ((float4*)boxes)[(size_t)b * PRE_NMS + kq] = box;
    }
}

// ---------------------------------------------------------------- 5) suppression bitmask via WMMA-assisted IoU
// 256-thread blocks = 8 waves; each wave owns one (16-row x 32-col) tile.
// union = area_i + area_j - inter via V_WMMA_F32_16X16X4_F32 with
// A=[area_row,1], B=[1;area_col], C=-inter. Threshold test is division-free:
// inter/(union+eps) > THR  <=>  inter > THR*(union+eps).
__global__ void __launch_bounds__(256)
nms_mask_kernel(const float* __restrict__ boxes, unsigned int* __restrict__ mask) {
    const int lane = threadIdx.x & 31;           // wave32 lane
    const int wave = threadIdx.x >> 5;           // 0..7
    const int w    = blockIdx.x * 8 + wave;      // col word 0..63 (32 cols each)
    const int rt   = blockIdx.y;                 // row tile 0..127 (16 rows each)
    const int b    = blockIdx.z;
    const float4* bx = (const float4*)boxes + (size_t)b * PRE_NMS;
    const int r0   = rt * 16;
    const int c0   = w * 32;
    const int nsub = lane & 15;
    const int grp  = lane >> 4;                  // 0: M=v, 1: M=v+8

    // strictly-lower-triangle tiles contain only zeros (need j > i)
    if (c0 + 31 <= r0) {
        if (lane < 16)
            mask[((size_t)b * PRE_NMS + (r0 + lane)) * 64 + w] = 0u;
        return;                                  // wave-uniform branch
    }

    // row boxes for accumulator rows M = v + 8*grp
    float4 rb[8];
    #pragma unroll
    for (int v = 0; v < 8; ++v) rb[v] = bx[r0 + v + 8 * grp];
    // A-operand row area: lanes 0-15 hold rows M=lane (K=0), K=1 is the constant 1
    float4 rba = bx[r0 + nsub];
    float areaRow = (rba.z - rba.x) * (rba.w - rba.y);

    unsigned int baA[8], baB[8];
    #pragma unroll
    for (int half = 0; half < 2; ++half) {
        const int c = c0 + half * 16 + nsub;     // column for this lane (N = lane&15)
        float4 cb = bx[c];
        float areaCol = (cb.z - cb.x) * (cb.w - cb.y);

        v8f cacc; float inter[8];
        #pragma unroll
        for (int v = 0; v < 8; ++v) {
            float ih = fminf(rb[v].z, cb.z) - fmaxf(rb[v].x, cb.x); ih = fmaxf(ih, 0.f);
            float iw = fminf(rb[v].w, cb.w) - fmaxf(rb[v].y, cb.y); iw = fmaxf(iw, 0.f);
            float in = ih * iw;
            inter[v] = in;
            cacc[v]  = -in;
        }
        // A 16x4: lanes 0-15 = {area_row, 1} (K0,K1); lanes 16-31 = K2,K3 = 0
        // B 4x16: lanes 0-15 = {1, area_col} (K0,K1); lanes 16-31 = 0
        v2f aop, bop;
        aop.x = (grp == 0) ? areaRow : 0.f;
        aop.y = (grp == 0) ? 1.f     : 0.f;
        bop.x = (grp == 0) ? 1.f     : 0.f;
        bop.y = (grp == 0) ? areaCol : 0.f;
        v8f un = __builtin_amdgcn_wmma_f32_16x16x4_f32(
            false, aop, false, bop, (short)0, cacc, false, false);

        #pragma unroll
        for (int v = 0; v < 8; ++v) {
            int i = r0 + v + 8 * grp;            // row index (matches D layout)
            // division-free IoU threshold; direct wave32 ballot (no cndmask round-trip)
            unsigned int ba = __builtin_amdgcn_ballot_w32(
                (inter[v] > NMS_THR * (un[v] + 1e-8f)) && (c > i));
            if (half == 0) baA[v] = ba; else baB[v] = ba;
        }
    }
    // ballot layout per v: bits[15:0] = row v cols 0-15, bits[31:16] = row v+8 cols 0-15
    if (lane < 16) {
        int v = lane & 7;
        unsigned int a = 0, bb2 = 0;
        #pragma unroll
        for (int q = 0; q < 8; ++q) if (q == v) { a = baA[q]; bb2 = baB[q]; }
        unsigned int word = (lane < 8)
            ? ((a & 0xFFFFu) | (bb2 << 16))
            : ((a >> 16)     | (bb2 & 0xFFFF0000u));
        mask[((size_t)b * PRE_NMS + (r0 + lane)) * 64 + w] = word;
    }
}

// ---------------------------------------------------------------- 6) sequential mask reduction + output
__global__ void nms_reduce_kernel(const float* __restrict__ boxes,
                                  const unsigned int* __restrict__ mask,
                                  float* __restrict__ out) {
    __shared__ unsigned int removed[64];
    __shared__ int kcount;
    const int b = blockIdx.x;
    const int tid = threadIdx.x;  // blockDim = 64
    if (tid < 64) removed[tid] = 0u;
    if (tid == 0) kcount = 0;
    for (int k = tid; k < NPROP * 4; k += blockDim.x)
        out[(size_t)b * NPROP * 4 + k] = 0.f;
    __syncthreads();
    for (int i = 0; i < PRE_NMS; ++i) {
        if (kcount >= NPROP) break;                              // uniform
        bool kept = ((removed[i >> 5] >> (i & 31)) & 1u) == 0u;  // uniform
        if (kept) {
            int kc = kcount;
            if (kc < NPROP && tid < 4)
                out[((size_t)b * NPROP + kc) * 4 + tid] =
                    boxes[((size_t)b * PRE_NMS + i) * 4 + tid];
            if (tid < 64) {
                __builtin_prefetch(&mask[((size_t)b * PRE_NMS + i + 1) * 64 + tid], 0, 1);
                removed[tid] |= mask[((size_t)b * PRE_NMS + i) * 64 + tid];
            }
            __syncthreads();
            if (tid == 0) kcount++;
        }
        __syncthreads();
    }
}

// ---------------------------------------------------------------- launcher
extern "C" void kernel_launch(void* const* d_in, const int* in_sizes, int n_in,
                              void* d_out, int out_size, void* d_ws, size_t ws_size,
                              hipStream_t stream) {
    const float* probs = (const float*)d_in[0];  // (B, N, 2)
    const float* bbox  = (const float*)d_in[1];  // (B, N, 4)
    const float* anch  = (const float*)d_in[2];  // (B, N, 4)
    const int N = in_sizes[0] / (B_ * 2);

    char* ws = (char*)d_ws;
    size_t off = 0;
    auto alloc = [&](size_t bytes) -> void* {
        void* p = ws + off;
        off = (off + bytes + 255) & ~(size_t)255;
        return p;
    };
    unsigned int* hist  = (unsigned int*)alloc((size_t)B_ * NBINS * 4);
    int*          tbin  = (int*)         alloc((size_t)B_ * 4);
    unsigned int* cnt   = (unsigned int*)alloc((size_t)B_ * 4);
    float*        cs    = (float*)       alloc((size_t)B_ * CAND_CAP * 4);
    int*          ci    = (int*)         alloc((size_t)B_ * CAND_CAP * 4);
    float*        boxes = (float*)       alloc((size_t)B_ * PRE_NMS * 4 * 4);
    unsigned int* mask  = (unsigned int*)alloc((size_t)B_ * PRE_NMS * 64 * 4);

    zero_ws<<<dim3(64), 256, 0, stream>>>(hist, B_ * NBINS);
    zero_ws<<<1, 64, 0, stream>>>(cnt, B_);
    hist_kernel<<<dim3(256, B_), 256, 0, stream>>>(probs, hist, N);
    thresh_kernel<<<1, 64, 0, stream>>>(hist, tbin);
    compact_kernel<<<dim3(256, B_), 256, 0, stream>>>(probs, tbin, cs, ci, cnt, N);
    sort_build_kernel<<<B_, 1024, 0, stream>>>(cs, ci, cnt, bbox, anch, boxes, N);
    nms_mask_kernel<<<dim3(8, 128, B_), 256, 0, stream>>>(boxes, mask);
    nms_reduce_kernel<<<B_, 64, 0, stream>>>(boxes, mask, (float*)d_out);
}